// UnsupervisedGAT_8942121910874
// MI455X (gfx1250) — compile-verified
//
#include <hip/hip_runtime.h>
#include <cstddef>

typedef __attribute__((ext_vector_type(2))) float v2f;
typedef __attribute__((ext_vector_type(8))) float v8f;

#define HD    256   // H*D hidden width
#define NHEAD 8

__device__ __forceinline__ float lrelu(float v, float s) { return v > 0.0f ? v : s * v; }

// ---------------------------------------------------------------------------
// C[M x 256] = A[M x K] @ B[K x 256] in f32 via V_WMMA_F32_16X16X4_F32.
// Block = 128 threads (4 waves). Wave w computes rows [16*bx .. +16) x cols [64*w .. +64)
// as four 16x16 tiles sharing one A operand per K-step (A reuse x4).
// ISA 7.12.2 layouts:
//   A 16x4 : lane<16 -> M=lane,   {K=k,k+1};  lane>=16 -> M=lane-16, {K=k+2,k+3}
//   B 4x16 : lane<16 -> N=lane,   {K=k,k+1};  lane>=16 -> N=lane-16, {K=k+2,k+3}
//   C/D    : VGPR j  : lane<16 -> M=j, N=lane; lane>=16 -> M=j+8, N=lane-16
// ---------------------------------------------------------------------------
__global__ void __launch_bounds__(128) gat_gemm(const float* __restrict__ A,
                                                const float* __restrict__ B,
                                                float* __restrict__ C, int K) {
  const int lane = threadIdx.x & 31;
  const int wave = threadIdx.x >> 5;
  const int row0 = blockIdx.x << 4;
  const int col0 = wave << 6;
  const int lh   = lane & 15;
  const int hi   = lane >> 4;   // 0: lanes 0-15, 1: lanes 16-31
  const int kk   = hi << 1;     // K sub-offset within a 4-step: 0 or 2

  v8f acc0 = {}, acc1 = {}, acc2 = {}, acc3 = {};
  const float* arow = A + (size_t)(row0 + lh) * K + kk;
  for (int k = 0; k < K; k += 4) {
    v2f a = *(const v2f*)(arow + k);                 // contiguous 8B -> global_load_b64
    const float* bp = B + (size_t)(k + kk) * HD + col0 + lh;
    v2f b0, b1, b2, b3;
    b0.x = bp[0];  b0.y = bp[HD];
    b1.x = bp[16]; b1.y = bp[HD + 16];
    b2.x = bp[32]; b2.y = bp[HD + 32];
    b3.x = bp[48]; b3.y = bp[HD + 48];
    // Same A operand for all 4 -> reuse_a hint on the first three.
    acc0 = __builtin_amdgcn_wmma_f32_16x16x4_f32(false, a, false, b0, (short)0, acc0, true,  false);
    acc1 = __builtin_amdgcn_wmma_f32_16x16x4_f32(false, a, false, b1, (short)0, acc1, true,  false);
    acc2 = __builtin_amdgcn_wmma_f32_16x16x4_f32(false, a, false, b2, (short)0, acc2, true,  false);
    acc3 = __builtin_amdgcn_wmma_f32_16x16x4_f32(false, a, false, b3, (short)0, acc3, false, false);
  }
  const int rbase = row0 + (hi << 3);
#pragma unroll
  for (int j = 0; j < 8; ++j) {
    float* cp = C + (size_t)(rbase + j) * HD + col0 + lh;
    cp[0]  = acc0[j];
    cp[16] = acc1[j];
    cp[32] = acc2[j];
    cp[48] = acc3[j];
  }
}

// ---------------------------------------------------------------------------
// el[n,h] = sum_d h[n,h,d]*al[h,d]; er likewise. One block (256 thr) per node.
// head == wave id (wave32), so each head reduces fully in-wave.
// ---------------------------------------------------------------------------
__global__ void __launch_bounds__(256) gat_elr(const float* __restrict__ h,
                                               const float* __restrict__ al,
                                               const float* __restrict__ ar,
                                               float* __restrict__ el,
                                               float* __restrict__ er) {
  const int n = blockIdx.x;
  const int c = threadIdx.x;                  // 0..255 -> head = c>>5, d = c&31
  float hv = h[(size_t)n * HD + c];
  float pl = hv * al[c];
  float pr = hv * ar[c];
#pragma unroll
  for (int off = 16; off > 0; off >>= 1) {
    pl += __shfl_xor(pl, off, 32);
    pr += __shfl_xor(pr, off, 32);
  }
  if ((c & 31) == 0) {
    el[n * NHEAD + (c >> 5)] = pl;
    er[n * NHEAD + (c >> 5)] = pr;
  }
}

// Zero the accumulation buffer; init per-node max to -inf and denom to 0.
__global__ void __launch_bounds__(256) gat_init(float* __restrict__ acc,
                                                float* __restrict__ mx,
                                                float* __restrict__ denom, int Nn) {
  int i = blockIdx.x * 256 + threadIdx.x;
  if (i < Nn * HD) acc[i] = 0.0f;
  if (i < Nn * NHEAD) {
    mx[i]    = -__builtin_inff();
    denom[i] = 0.0f;
  }
}

// Pass 1: per-edge score -> segment max over dst (global_atomic_max_num_f32).
__global__ void __launch_bounds__(256) gat_edge_max(const int* __restrict__ src,
                                                    const int* __restrict__ dst,
                                                    const float* __restrict__ el,
                                                    const float* __restrict__ er,
                                                    float* __restrict__ mx, int E) {
  int t = blockIdx.x * 256 + threadIdx.x;
  if (t >= E * NHEAD) return;
  int e  = t >> 3, hh = t & 7;
  int s  = src[e], d = dst[e];
  float v = lrelu(el[s * NHEAD + hh] + er[d * NHEAD + hh], 0.2f);
  atomicMax(&mx[d * NHEAD + hh], v);
}

// Pass 2: denom[dst,h] += exp(score - m[dst,h])
__global__ void __launch_bounds__(256) gat_edge_denom(const int* __restrict__ src,
                                                      const int* __restrict__ dst,
                                                      const float* __restrict__ el,
                                                      const float* __restrict__ er,
                                                      const float* __restrict__ mx,
                                                      float* __restrict__ denom, int E) {
  int t = blockIdx.x * 256 + threadIdx.x;
  if (t >= E * NHEAD) return;
  int e  = t >> 3, hh = t & 7;
  int s  = src[e], d = dst[e];
  int di = d * NHEAD + hh;
  float v = lrelu(el[s * NHEAD + hh] + er[d * NHEAD + hh], 0.2f);
  atomicAdd(&denom[di], __expf(v - mx[di]));
}

// Pass 3: out[dst,c] += a[e,head(c)] * h[src,c]. One block (256 thr) per edge.
__global__ void __launch_bounds__(256) gat_edge_agg(const int* __restrict__ src,
                                                    const int* __restrict__ dst,
                                                    const float* __restrict__ h,
                                                    const float* __restrict__ el,
                                                    const float* __restrict__ er,
                                                    const float* __restrict__ mx,
                                                    const float* __restrict__ denom,
                                                    float* __restrict__ out, int E) {
  const int e  = blockIdx.x;
  const int c  = threadIdx.x;          // channel 0..255
  const int hh = c >> 5;               // head (uniform per wave -> scalar loads)
  const int s  = src[e], d = dst[e];
  const int di = d * NHEAD + hh;
  float v = lrelu(el[s * NHEAD + hh] + er[di], 0.2f);
  float a = __expf(v - mx[di]) / fmaxf(denom[di], 1e-9f);
  atomicAdd(&out[(size_t)d * HD + c], h[(size_t)s * HD + c] * a);
}

// In-place leaky_relu(x, 0.01) after layers 0 and 1.
__global__ void __launch_bounds__(256) gat_act(float* __restrict__ x, int Nn) {
  int i = blockIdx.x * 256 + threadIdx.x;
  if (i < Nn * HD) x[i] = lrelu(x[i], 0.01f);
}

// ---------------------------------------------------------------------------
extern "C" void kernel_launch(void* const* d_in, const int* in_sizes, int n_in,
                              void* d_out, int out_size, void* d_ws, size_t ws_size,
                              hipStream_t stream) {
  const float* n_feat = (const float*)d_in[0];
  // d_in[1] (e_feat) is unused by the reference layers.
  const float* W0     = (const float*)d_in[2];   // [128, 256]
  const float* Ws     = (const float*)d_in[3];   // [2, 256, 256]
  const float* attn_l = (const float*)d_in[4];   // [3, 8, 32]
  const float* attn_r = (const float*)d_in[5];   // [3, 8, 32]
  const int*   eidx   = (const int*)d_in[6];     // [2, E]

  const int IN_DIM = 128;
  const int Nn = in_sizes[0] / IN_DIM;           // 50000
  const int E  = in_sizes[6] / 2;                // 800000
  const int* src = eidx;
  const int* dst = eidx + E;

  float* ws    = (float*)d_ws;
  float* xbuf  = ws;                                  // N*256  (layer output / next input)
  float* hbuf  = xbuf + (size_t)Nn * HD;              // N*256  (h = x @ W)
  float* el    = hbuf + (size_t)Nn * HD;              // N*8
  float* er    = el + (size_t)Nn * NHEAD;             // N*8
  float* mx    = er + (size_t)Nn * NHEAD;             // N*8
  float* denom = mx + (size_t)Nn * NHEAD;             // N*8

  const int nodeBlk = (Nn * HD + 255) / 256;
  const int edgeBlk = (E * NHEAD + 255) / 256;

  for (int layer = 0; layer < 3; ++layer) {
    const float* A = (layer == 0) ? n_feat : xbuf;
    const int    K = (layer == 0) ? IN_DIM : HD;
    const float* W = (layer == 0) ? W0 : Ws + (size_t)(layer - 1) * HD * HD;
    float* acc = (layer == 2) ? (float*)d_out : xbuf;  // xbuf is dead after the GEMM
    const float* al = attn_l + layer * HD;
    const float* ar = attn_r + layer * HD;

    gat_gemm<<<Nn / 16, 128, 0, stream>>>(A, W, hbuf, K);
    gat_elr<<<Nn, HD, 0, stream>>>(hbuf, al, ar, el, er);
    gat_init<<<nodeBlk, 256, 0, stream>>>(acc, mx, denom, Nn);
    gat_edge_max<<<edgeBlk, 256, 0, stream>>>(src, dst, el, er, mx, E);
    gat_edge_denom<<<edgeBlk, 256, 0, stream>>>(src, dst, el, er, mx, denom, E);
    gat_edge_agg<<<E, HD, 0, stream>>>(src, dst, hbuf, el, er, mx, denom, acc, E);
    if (layer < 2) gat_act<<<nodeBlk, 256, 0, stream>>>(xbuf, Nn);
  }
}